// RoIPool2D_52441550684761
// MI455X (gfx1250) — compile-verified
//
#include <hip/hip_runtime.h>
#include <stdint.h>

#define POOL_H 7
#define POOL_W 7
#define KH_    10
#define KW_    10
#define N_     4
#define C_     128
#define H_     50
#define W_     50
#define R_     256
#define SCALE_ 0.0625f
#define NBINS  (POOL_H * POOL_W)          // 49
#define PER_ROI (C_ * NBINS)              // 6272 floats per roi

// ---------------------------------------------------------------------------
// Kernel 1: NCHW -> NHWC transpose of the feature map (into d_ws).
// Each thread produces one float4 of the NHWC output (4 consecutive channels
// at one (b,y,x)) -> b128 coalesced stores; the 4 strided reads are
// L2-resident after first touch (feat is only 5.12 MB vs 192 MB L2).
// ---------------------------------------------------------------------------
__global__ void nchw_to_nhwc_kernel(const float* __restrict__ in,
                                    float* __restrict__ out) {
    const int total4 = (N_ * C_ * H_ * W_) / 4;          // 320000
    int idx4 = blockIdx.x * blockDim.x + threadIdx.x;    // float4 index (NHWC)
    if (idx4 >= total4) return;
    int c  = (idx4 << 2) & (C_ - 1);                     // 0,4,...,124
    int t  = idx4 >> 5;                                  // b*H*W + yx
    int yx = t % (H_ * W_);
    int b  = t / (H_ * W_);
    const float* src = in + ((size_t)(b * C_ + c)) * (H_ * W_) + yx;
    float4 v;
    v.x = src[0 * H_ * W_];
    v.y = src[1 * H_ * W_];
    v.z = src[2 * H_ * W_];
    v.w = src[3 * H_ * W_];
    *(float4*)(out + ((size_t)idx4 << 2)) = v;
}

// ---------------------------------------------------------------------------
// Shared helper: per-RoI bin boundaries, matching the reference exactly.
// rintf under default RTE mode == jnp.round (round half to even).
// ---------------------------------------------------------------------------
__device__ __forceinline__ void compute_bins(const float* roi, int tid,
                                             int* s_hs, int* s_he,
                                             int* s_ws, int* s_we, int& b) {
    b = (int)roi[0];
    const int   xs = (int)rintf(roi[1] * SCALE_);
    const int   ys = (int)rintf(roi[2] * SCALE_);
    const int   xe = (int)rintf(roi[3] * SCALE_);
    const int   ye = (int)rintf(roi[4] * SCALE_);
    const float bin_h = (float)max(ye - ys + 1, 1) / (float)POOL_H;
    const float bin_w = (float)max(xe - xs + 1, 1) / (float)POOL_W;
    if (tid < POOL_H) {
        int hs = (int)floorf((float)tid * bin_h) + ys;
        int he = (int)ceilf(((float)tid + 1.0f) * bin_h) + ys;
        s_hs[tid] = min(max(hs, 0), H_);
        s_he[tid] = min(max(he, 0), H_);
    } else if (tid < POOL_H + POOL_W) {
        int pw = tid - POOL_H;
        int w0 = (int)floorf((float)pw * bin_w) + xs;
        int w1 = (int)ceilf(((float)pw + 1.0f) * bin_w) + xs;
        s_ws[pw] = min(max(w0, 0), W_);
        s_we[pw] = min(max(w1, 0), W_);
    }
}

// ---------------------------------------------------------------------------
// Shared helper: async DMA the staged 25088-byte LDS tile to global memory.
// CDNA5 path: global_store_async_from_lds_b128 (16B/lane, fully coalesced,
// no VGPR round-trip), tracked by ASYNCcnt, then s_wait_asynccnt 0.
// ---------------------------------------------------------------------------
__device__ __forceinline__ void async_writeout(const float* pool, float* dst,
                                               int tid, int nthreads) {
    const uint64_t dst_base = (uint64_t)(uintptr_t)dst;
    const uint32_t lds_base = (uint32_t)(uintptr_t)pool;  // low 32b = LDS offset
    const int nvec = PER_ROI / 4;                         // 1568 x float4
    for (int i4 = tid; i4 < nvec; i4 += nthreads) {
        uint32_t byte_off = (uint32_t)i4 * 16u;
        uint32_t laddr    = lds_base + byte_off;
        asm volatile("global_store_async_from_lds_b128 %0, %1, %2"
                     :: "v"(byte_off), "v"(laddr), "s"(dst_base)
                     : "memory");
    }
#if __has_builtin(__builtin_amdgcn_s_wait_asynccnt)
    __builtin_amdgcn_s_wait_asynccnt(0);
#else
    asm volatile("s_wait_asynccnt 0x0" ::: "memory");
#endif
}

// ---------------------------------------------------------------------------
// Kernel 2 (fast path, NHWC): one workgroup (256 threads, 8 wave32) per RoI.
// Each thread owns (bin, 4-channel group): float4 gather loads -> one wave32
// covers all 128 channels of a bin with a single contiguous 512B transaction
// per (y,x). Results staged in LDS in final (c,ph,pw) order, then DMA'd out.
// ---------------------------------------------------------------------------
__global__ void roipool_nhwc_kernel(const float* __restrict__ feat,
                                    const float* __restrict__ rois,
                                    float* __restrict__ out) {
    __shared__ int s_hs[POOL_H], s_he[POOL_H];
    __shared__ int s_ws[POOL_W], s_we[POOL_W];
    __shared__ __attribute__((aligned(16))) float pool[PER_ROI];

    const int r = blockIdx.x;
    const int tid = threadIdx.x;
    int b;
    compute_bins(rois + (size_t)r * 5, tid, s_hs, s_he, s_ws, s_we, b);
    __syncthreads();

    for (int t = tid; t < PER_ROI / 4; t += blockDim.x) {
        const int c4  = t & 31;                  // channels c4*4 .. c4*4+3
        const int bin = t >> 5;                  // 0..48
        const int ph  = bin / POOL_W;
        const int pw  = bin - ph * POOL_W;

        const int hs = s_hs[ph], he = s_he[ph];
        const int w0 = s_ws[pw], w1 = s_we[pw];
        const int hc = min(he - hs, KH_);
        const int wc = min(w1 - w0, KW_);

        float4 acc;
        if (hc > 0 && wc > 0) {
            acc.x = acc.y = acc.z = acc.w = -3.4e38f;
            const float* rowp =
                feat + ((size_t)((b * H_ + hs) * W_ + w0)) * C_ + (c4 << 2);
            for (int iy = 0; iy < hc; ++iy) {
                if (iy + 1 < hc)                         // row-ahead prefetch
                    __builtin_prefetch(rowp + W_ * C_, 0, 0); // global_prefetch_b8
                for (int ix = 0; ix < wc; ++ix) {
                    float4 v = *(const float4*)(rowp + ix * C_);
                    acc.x = fmaxf(acc.x, v.x);
                    acc.y = fmaxf(acc.y, v.y);
                    acc.z = fmaxf(acc.z, v.z);
                    acc.w = fmaxf(acc.w, v.w);
                }
                rowp += W_ * C_;
            }
        } else {
            acc.x = acc.y = acc.z = acc.w = 0.0f;        // empty bin
        }
        const int base = (c4 << 2) * NBINS + bin;        // final output order
        pool[base + 0 * NBINS] = acc.x;
        pool[base + 1 * NBINS] = acc.y;
        pool[base + 2 * NBINS] = acc.z;
        pool[base + 3 * NBINS] = acc.w;
    }
    __syncthreads();
    async_writeout(pool, out + (size_t)r * PER_ROI, tid, blockDim.x);
}

// ---------------------------------------------------------------------------
// Kernel 2 (fallback, NCHW direct): scalar per-channel gather (channel stride
// H*W prevents vectorization). Used only if d_ws is too small.
// ---------------------------------------------------------------------------
__global__ void roipool_nchw_kernel(const float* __restrict__ feat,
                                    const float* __restrict__ rois,
                                    float* __restrict__ out) {
    __shared__ int s_hs[POOL_H], s_he[POOL_H];
    __shared__ int s_ws[POOL_W], s_we[POOL_W];
    __shared__ __attribute__((aligned(16))) float pool[PER_ROI];

    const int r = blockIdx.x;
    const int tid = threadIdx.x;
    int b;
    compute_bins(rois + (size_t)r * 5, tid, s_hs, s_he, s_ws, s_we, b);
    __syncthreads();

    for (int t = tid; t < PER_ROI; t += blockDim.x) {
        const int c   = t & (C_ - 1);
        const int bin = t >> 7;
        const int ph  = bin / POOL_W;
        const int pw  = bin - ph * POOL_W;

        const int hs = s_hs[ph], he = s_he[ph];
        const int w0 = s_ws[pw], w1 = s_we[pw];
        const int hc = min(he - hs, KH_);
        const int wc = min(w1 - w0, KW_);

        float acc;
        if (hc > 0 && wc > 0) {
            acc = -3.4e38f;
            const float* rowp = feat + ((size_t)(b * C_ + c) * H_ + hs) * W_ + w0;
            for (int iy = 0; iy < hc; ++iy) {
                if (iy + 1 < hc)
                    __builtin_prefetch(rowp + W_, 0, 0);
                for (int ix = 0; ix < wc; ++ix)
                    acc = fmaxf(acc, rowp[ix]);
                rowp += W_;
            }
        } else {
            acc = 0.0f;
        }
        pool[c * NBINS + bin] = acc;
    }
    __syncthreads();
    async_writeout(pool, out + (size_t)r * PER_ROI, tid, blockDim.x);
}

// ---------------------------------------------------------------------------
extern "C" void kernel_launch(void* const* d_in, const int* in_sizes, int n_in,
                              void* d_out, int out_size, void* d_ws, size_t ws_size,
                              hipStream_t stream) {
    (void)in_sizes; (void)n_in; (void)out_size;
    const float* feat = (const float*)d_in[0];   // (4,128,50,50) f32
    const float* rois = (const float*)d_in[1];   // (256,5) f32
    float* out = (float*)d_out;                  // (256,128,7,7) f32

    const size_t featT_bytes = (size_t)N_ * C_ * H_ * W_ * sizeof(float);
    if (d_ws != nullptr && ws_size >= featT_bytes) {
        float* featT = (float*)d_ws;
        const int total4 = (N_ * C_ * H_ * W_) / 4;
        nchw_to_nhwc_kernel<<<(total4 + 255) / 256, 256, 0, stream>>>(feat, featT);
        roipool_nhwc_kernel<<<R_, 256, 0, stream>>>(featT, rois, out);
    } else {
        roipool_nchw_kernel<<<R_, 256, 0, stream>>>(feat, rois, out);
    }
}